// DeepGCN_63015760166991
// MI455X (gfx1250) — compile-verified
//
#include <hip/hip_runtime.h>

// DeepGCN on MI455X (gfx1250, wave32).
// GEMMs via V_WMMA_F32_16X16X4_F32 (f32 precision, matrix pipe), branch-free
// inner loop (weights always a dense FINx16 tile; last layer zero-padded).
// Edge scatter via float4 gathers + native global_atomic_add_f32; hw/agg
// buffers (6.4MB each) and the edge stream (~64MB) stay resident in the
// 192MB L2, so HBM traffic is minimal after the first pass.

typedef float v2f __attribute__((ext_vector_type(2)));
typedef float v8f __attribute__((ext_vector_type(8)));

static constexpr int D_IN  = 128;
static constexpr int HDIM  = 16;
static constexpr int D_OUT = 12;
static constexpr int N_MID = 11;

__device__ __forceinline__ void atomAddF(float* p, float v) {
  unsafeAtomicAdd(p, v);   // lowers to GLOBAL_ATOMIC_ADD_F32 (no-return)
}

// ---------------- precompute: degree, dis=rsqrt(deg), self_w=1/deg, norm ---

__global__ void zero_f32_kernel(float* __restrict__ p, int n) {
  int t = blockIdx.x * blockDim.x + threadIdx.x;
  if (t < n) p[t] = 0.0f;
}

__global__ void deg_scatter_kernel(const long long* __restrict__ col,
                                   const float* __restrict__ ew,
                                   float* __restrict__ deg, int E) {
  int e = blockIdx.x * blockDim.x + threadIdx.x;
  if (e < E) atomAddF(&deg[col[e]], ew[e]);
}

// in: deg (sum of incoming weights); out (in-place): dis = rsqrt(deg+1), self_w = 1/(deg+1)
__global__ void deg_finalize_kernel(float* __restrict__ deg_dis,
                                    float* __restrict__ self_w, int n) {
  int t = blockIdx.x * blockDim.x + threadIdx.x;
  if (t < n) {
    float d = deg_dis[t] + 1.0f;
    deg_dis[t] = __frsqrt_rn(d);
    self_w[t]  = 1.0f / d;
  }
}

__global__ void norm_kernel(const long long* __restrict__ row,
                            const long long* __restrict__ col,
                            const float* __restrict__ ew,
                            const float* __restrict__ dis,
                            float* __restrict__ norm, int E) {
  int e = blockIdx.x * blockDim.x + threadIdx.x;
  if (e < E) norm[e] = dis[row[e]] * ew[e] * dis[col[e]];
}

// ---- pad last-layer weight (16x12 -> 16x16) and bias (12 -> 16) with zeros -

__global__ void pad_last_kernel(const float* __restrict__ Wlast,
                                const float* __restrict__ blast,
                                float* __restrict__ Wpad,
                                float* __restrict__ bpad) {
  int t = threadIdx.x;            // single block of 256
  int k = t >> 4, n = t & 15;
  Wpad[t] = (n < D_OUT) ? Wlast[k * D_OUT + n] : 0.0f;
  if (t < HDIM) bpad[t] = (t < D_OUT) ? blast[t] : 0.0f;
}

// ---------------- GEMM: HW[N x 16] = Hm[N x FIN] @ Wm[FIN x 16] -------------

// One wave (32 lanes) per 16-row output tile. FIN is compile-time so the
// K loop fully unrolls with zero control flow around the WMMAs.
template <int FIN>
__global__ void gemm_wmma_f32_kernel(const float* __restrict__ Hm,
                                     const float* __restrict__ Wm,
                                     float* __restrict__ HW, int nrows) {
  const int wave = (int)((blockIdx.x * blockDim.x + threadIdx.x) >> 5);
  const int lane = (int)(threadIdx.x & 31);
  const int m0   = wave * 16;
  if (m0 >= nrows) return;                 // wave-uniform: EXEC stays all-ones
  const int r  = lane & 15;                // A row-in-tile / B,C,D column
  const int kh = (lane >> 4) << 1;         // K sub-offset: 0 or 2
  v8f acc = {};
  const float* arow = Hm + (size_t)(m0 + r) * FIN;
#pragma unroll
  for (int k0 = 0; k0 < FIN; k0 += 4) {
    v2f a, b;
    // A 16x4 f32 layout: lanes 0-15 hold K=k0,k0+1; lanes 16-31 hold K=k0+2,k0+3
    a.x = arow[k0 + kh];
    a.y = arow[k0 + kh + 1];
    // B 4x16 f32 layout: VGPR0 rows K=k0 (lanes 0-15) / K=k0+2 (lanes 16-31)
    b.x = Wm[(k0 + kh) * 16 + r];
    b.y = Wm[(k0 + kh + 1) * 16 + r];
    acc = __builtin_amdgcn_wmma_f32_16x16x4_f32(false, a, false, b,
                                                (short)0, acc, false, false);
  }
  // C/D 16x16 f32: VGPR v -> M = v + (lane<16 ? 0 : 8), N = lane&15
  const int mo = (lane >> 4) << 3;
  float* orow = HW + ((size_t)(m0 + mo)) * 16 + r;
#pragma unroll
  for (int v = 0; v < 8; ++v) orow[(size_t)v * 16] = acc[v];
}

// ---------------- agg init: self-loop term + bias (bias always 16-wide) -----

__global__ void init_agg_kernel(const float* __restrict__ hw,
                                const float* __restrict__ self_w,
                                const float* __restrict__ bias,
                                float* __restrict__ agg, int n) {
  int t = blockIdx.x * blockDim.x + threadIdx.x;
  if (t < n * 16) {
    int i = t >> 4, f = t & 15;
    agg[t] = self_w[i] * hw[t] + bias[f];
  }
}

// ---------------- edge scatter: agg[col] += norm[e] * hw[row] ---------------

// 4 lanes per edge; each lane handles a float4 slice of the 16 features, so
// a 4-lane group reads one coalesced 64B line and issues 4 L2 atomics each.
__global__ void edge_scatter_kernel(const long long* __restrict__ row,
                                    const long long* __restrict__ col,
                                    const float* __restrict__ norm,
                                    const float* __restrict__ hw,
                                    float* __restrict__ agg, int E) {
  long long t = (long long)blockIdx.x * blockDim.x + threadIdx.x;
  int e   = (int)(t >> 2);
  int sub = (int)(t & 3);
  if (e >= E) return;
  long long rs = row[e];
  long long cd = col[e];
  float w = norm[e];
  const float4 v = *reinterpret_cast<const float4*>(hw + rs * 16 + sub * 4);
  float* dst = agg + cd * 16 + sub * 4;
  atomAddF(dst + 0, w * v.x);
  atomAddF(dst + 1, w * v.y);
  atomAddF(dst + 2, w * v.z);
  atomAddF(dst + 3, w * v.w);
}

// ---------------- relu (agg -> h) and final packed output -------------------

__global__ void relu_kernel(const float* __restrict__ agg,
                            float* __restrict__ h, int n16) {
  int t = blockIdx.x * blockDim.x + threadIdx.x;
  if (t < n16) h[t] = fmaxf(agg[t], 0.0f);
}

__global__ void final_out_kernel(const float* __restrict__ agg,
                                 float* __restrict__ out, int n) {
  int t = blockIdx.x * blockDim.x + threadIdx.x;   // over n * D_OUT
  if (t < n * D_OUT) {
    int i = t / D_OUT, f = t % D_OUT;
    out[t] = fmaxf(agg[i * 16 + f], 0.0f);
  }
}

// ---------------------------------------------------------------------------

extern "C" void kernel_launch(void* const* d_in, const int* in_sizes, int n_in,
                              void* d_out, int out_size, void* d_ws, size_t ws_size,
                              hipStream_t stream) {
  const float*     x     = (const float*)d_in[0];            // [N,128]
  const long long* eidx  = (const long long*)d_in[1];        // [2,E] int64
  const float*     ew    = (const float*)d_in[2];            // [E]
  const float*     W0    = (const float*)d_in[3];            // [128,16]
  const float*     b0    = (const float*)d_in[4];            // [16]
  const float*     Wmid  = (const float*)d_in[5];            // [11,16,16]
  const float*     bmid  = (const float*)d_in[6];            // [11,16]
  const float*     Wlast = (const float*)d_in[7];            // [16,12]
  const float*     blast = (const float*)d_in[8];            // [12]
  float* out = (float*)d_out;

  const int N = in_sizes[0] / D_IN;
  const int E = in_sizes[2];
  const long long* rowp = eidx;
  const long long* colp = eidx + E;

  // workspace layout (floats)
  float* ws     = (float*)d_ws;
  float* norm   = ws;                       // E
  float* dis    = norm + (size_t)E;         // N (degree accumulator, then rsqrt)
  float* selfw  = dis + (size_t)N;          // N
  float* hbuf   = selfw + (size_t)N;        // N*16
  float* hw     = hbuf + (size_t)N * 16;    // N*16
  float* agg    = hw + (size_t)N * 16;      // N*16
  float* Wpad   = agg + (size_t)N * 16;     // 256
  float* bpad   = Wpad + 256;               // 16

  const int TB = 256;
  auto blk = [](long long n, int tb) { return (unsigned)((n + tb - 1) / tb); };

  // ---- precompute norm / self weights / padded last layer (every call) -----
  zero_f32_kernel<<<blk(N, TB), TB, 0, stream>>>(dis, N);
  deg_scatter_kernel<<<blk(E, TB), TB, 0, stream>>>(colp, ew, dis, E);
  deg_finalize_kernel<<<blk(N, TB), TB, 0, stream>>>(dis, selfw, N);
  norm_kernel<<<blk(E, TB), TB, 0, stream>>>(rowp, colp, ew, dis, norm, E);
  pad_last_kernel<<<1, 256, 0, stream>>>(Wlast, blast, Wpad, bpad);

  const unsigned gemm_blocks = blk(((long long)N / 16) * 32, TB);  // 1 wave / 16-row tile
  const unsigned n16_blocks  = blk((long long)N * 16, TB);
  const unsigned edge_blocks = blk((long long)E * 4, TB);

  // ---- layer 0: 128 -> 16 ----
  gemm_wmma_f32_kernel<D_IN><<<gemm_blocks, TB, 0, stream>>>(x, W0, hw, N);
  init_agg_kernel<<<n16_blocks, TB, 0, stream>>>(hw, selfw, b0, agg, N);
  edge_scatter_kernel<<<edge_blocks, TB, 0, stream>>>(rowp, colp, norm, hw, agg, E);
  relu_kernel<<<n16_blocks, TB, 0, stream>>>(agg, hbuf, N * 16);

  // ---- 11 middle layers: 16 -> 16 ----
  for (int l = 0; l < N_MID; ++l) {
    const float* Wl = Wmid + (size_t)l * HDIM * HDIM;
    const float* bl = bmid + (size_t)l * HDIM;
    gemm_wmma_f32_kernel<HDIM><<<gemm_blocks, TB, 0, stream>>>(hbuf, Wl, hw, N);
    init_agg_kernel<<<n16_blocks, TB, 0, stream>>>(hw, selfw, bl, agg, N);
    edge_scatter_kernel<<<edge_blocks, TB, 0, stream>>>(rowp, colp, norm, hw, agg, E);
    relu_kernel<<<n16_blocks, TB, 0, stream>>>(agg, hbuf, N * 16);
  }

  // ---- last layer: 16 -> 12 (zero-padded to 16 internally) ----
  gemm_wmma_f32_kernel<HDIM><<<gemm_blocks, TB, 0, stream>>>(hbuf, Wpad, hw, N);
  init_agg_kernel<<<n16_blocks, TB, 0, stream>>>(hw, selfw, bpad, agg, N);
  edge_scatter_kernel<<<edge_blocks, TB, 0, stream>>>(rowp, colp, norm, hw, agg, E);
  final_out_kernel<<<blk((long long)N * D_OUT, TB), TB, 0, stream>>>(agg, out, N);
}